// AttnAutoEncoderRNN_14817637171431
// MI455X (gfx1250) — compile-verified
//
#include <hip/hip_runtime.h>
#include <hip/hip_bf16.h>
#include <math.h>

#define Hs   1024
#define H3   3072
#define Ts   256
#define Vs   50257
#define SOS_TOK 1
#define MTILE 4            // 16-row M tiles per wave sharing one B fragment

typedef __attribute__((ext_vector_type(2))) float v2f;
typedef __attribute__((ext_vector_type(8))) float v8f;

// ---------------------------------------------------------------------------
// WMMA f32 GEMM:  C[M,N] = A[M,K] @ W[N,K]^T + bias[N]
// One wave computes a 64x16 C stripe (4 stacked 16x16 tiles) so each B
// fragment feeds 4 V_WMMA_F32_16X16X4_F32 ops -> 4x less weight traffic.
// A-frag (16x4): lanes 0-15 -> K {k,k+1}, lanes 16-31 -> K {k+2,k+3}, M = lane&15.
// B-frag (4x16): same K split per half-wave, N = lane&15.
// C/D frag: lane gives N = lane&15, VGPR v gives M = v + (lane>>4)*8.
// ---------------------------------------------------------------------------
__global__ void __launch_bounds__(256)
wmma_gemm_bias_f32(const float* __restrict__ A,
                   const float* __restrict__ W,
                   const float* __restrict__ bias,
                   float* __restrict__ C,
                   int M, int N, int K)
{
    const int wave  = threadIdx.x >> 5;
    const int lane  = threadIdx.x & 31;
    const int tileN = (blockIdx.x * (blockDim.x >> 5) + wave) * 16;
    const int tileM = blockIdx.y * (16 * MTILE);
    if (tileN >= N) return;               // wave-uniform: EXEC stays all-ones

    const int half = lane >> 4;           // 0 -> K {k,k+1} ; 1 -> K {k+2,k+3}
    const int sub  = lane & 15;

    const int ncol = tileN + sub;
    const int ncl  = (ncol < N) ? ncol : (N - 1);   // clamp OOB weight rows

    const float* __restrict__ wrow = W + (size_t)ncl * K + 2 * half;
    const float* __restrict__ arow[MTILE];
    #pragma unroll
    for (int i = 0; i < MTILE; ++i)
        arow[i] = A + (size_t)(tileM + i * 16 + sub) * K + 2 * half;

    v8f acc[MTILE];
    #pragma unroll
    for (int i = 0; i < MTILE; ++i) acc[i] = (v8f){};

    for (int k = 0; k < K; k += 4) {
        if ((k & 31) == 0) {
            __builtin_prefetch(wrow + k + 128, 0, 1);   // global_prefetch_b8
        }
        v2f b;
        b.x = wrow[k];  b.y = wrow[k + 1];
        #pragma unroll
        for (int i = 0; i < MTILE; ++i) {
            v2f a;
            a.x = arow[i][k];  a.y = arow[i][k + 1];
            acc[i] = __builtin_amdgcn_wmma_f32_16x16x4_f32(
                         /*neg_a=*/false, a, /*neg_b=*/false, b,
                         /*c_mod=*/(short)0, acc[i],
                         /*reuse_a=*/false, /*reuse_b=*/false);
        }
    }

    if (ncol < N) {
        const float bv = bias ? bias[ncol] : 0.0f;
        #pragma unroll
        for (int i = 0; i < MTILE; ++i) {
            const int m0 = tileM + i * 16 + half * 8;
            #pragma unroll
            for (int v = 0; v < 8; ++v)
                C[(size_t)(m0 + v) * N + ncol] = acc[i][v] + bv;
        }
    }
}

__device__ __forceinline__ float sigm(float x) { return 1.0f / (1.0f + __expf(-x)); }

__device__ __forceinline__ float dot4(const float4 a, const float4 b)
{
    return a.x * b.x + a.y * b.y + a.z * b.z + a.w * b.w;
}

// ---------------------------------------------------------------------------
// Encoder scan: sequential GRU, one workgroup, h lives in LDS.
// gx[t, 3H] = x_t @ Wih^T + bih was precomputed by the WMMA GEMM.
// Thread tid owns hidden dim `tid`; float4 loads -> global_load_b128.
// ---------------------------------------------------------------------------
__global__ void __launch_bounds__(1024)
encoder_scan(const float* __restrict__ gx,
             const float* __restrict__ Whh,
             const float* __restrict__ bhh,
             float* __restrict__ enc_outs /* [T,H] */)
{
    __shared__ float h[Hs];
    const int tid = threadIdx.x;
    h[tid] = 0.0f;
    __syncthreads();

    const float4* __restrict__ whr = (const float4*)(Whh + (size_t)(tid)          * Hs);
    const float4* __restrict__ whz = (const float4*)(Whh + (size_t)(tid + Hs)     * Hs);
    const float4* __restrict__ whn = (const float4*)(Whh + (size_t)(tid + 2 * Hs) * Hs);
    const float br = bhh[tid], bz = bhh[tid + Hs], bn = bhh[tid + 2 * Hs];
    const float4* h4 = (const float4*)h;

    for (int t = 0; t < Ts; ++t) {
        float hr = br, hz = bz, hn = bn;
        for (int j = 0; j < Hs / 4; ++j) {
            const float4 hj = h4[j];
            hr += dot4(hj, whr[j]);
            hz += dot4(hj, whz[j]);
            hn += dot4(hj, whn[j]);
        }
        const float* g = gx + (size_t)t * H3;
        const float r = sigm(g[tid]          + hr);
        const float z = sigm(g[tid + Hs]     + hz);
        const float n = tanhf(g[tid + 2*Hs]  + r * hn);
        const float hnew = (1.0f - z) * n + z * h[tid];
        __syncthreads();
        h[tid] = hnew;
        enc_outs[(size_t)t * Hs + tid] = hnew;
        __syncthreads();
    }
}

// ---------------------------------------------------------------------------
// Decoder scan: Bahdanau attention + GRU, teacher forcing. One workgroup.
// Emits h_t to Hdec; the vocab projection is deferred to the batched WMMA GEMM.
// ---------------------------------------------------------------------------
__global__ void __launch_bounds__(1024)
decoder_scan(const int*   __restrict__ target,
             const float* __restrict__ emb_dec,
             const float* __restrict__ attn_W, const float* __restrict__ attn_b,
             const float* __restrict__ comb_W, const float* __restrict__ comb_b,
             const float* __restrict__ Wih, const float* __restrict__ bih,
             const float* __restrict__ Whh, const float* __restrict__ bhh,
             const float* __restrict__ enc_outs,
             float* __restrict__ Hdec /* [T,H] */)
{
    __shared__ float h[Hs], emb[Hs], ctx[Hs], comb[Hs];
    __shared__ float sc[Ts], red[Ts];
    const int tid = threadIdx.x;

    h[tid] = enc_outs[(size_t)(Ts - 1) * Hs + tid];   // h_enc (final encoder state)
    __syncthreads();

    const float4* __restrict__ wir = (const float4*)(Wih + (size_t)(tid)          * Hs);
    const float4* __restrict__ wiz = (const float4*)(Wih + (size_t)(tid + Hs)     * Hs);
    const float4* __restrict__ win = (const float4*)(Wih + (size_t)(tid + 2 * Hs) * Hs);
    const float4* __restrict__ whr = (const float4*)(Whh + (size_t)(tid)          * Hs);
    const float4* __restrict__ whz = (const float4*)(Whh + (size_t)(tid + Hs)     * Hs);
    const float4* __restrict__ whn = (const float4*)(Whh + (size_t)(tid + 2 * Hs) * Hs);
    const float bxr = bih[tid], bxz = bih[tid + Hs], bxn = bih[tid + 2 * Hs];
    const float bhr = bhh[tid], bhz = bhh[tid + Hs], bhn = bhh[tid + 2 * Hs];

    const float4* h4    = (const float4*)h;
    const float4* emb4  = (const float4*)emb;
    const float4* ctx4  = (const float4*)ctx;
    const float4* comb4 = (const float4*)comb;

    for (int t = 0; t < Ts; ++t) {
        const int tok = (t == 0) ? SOS_TOK : target[t - 1];
        emb[tid] = emb_dec[(size_t)tok * Hs + tid];
        __syncthreads();

        // attention scores over T: thread j<T does a 2H dot with attn_W[j]
        if (tid < Ts) {
            const float4* w = (const float4*)(attn_W + (size_t)tid * (2 * Hs));
            float s = attn_b[tid];
            for (int j = 0; j < Hs / 4; ++j) s += dot4(emb4[j], w[j]);
            for (int j = 0; j < Hs / 4; ++j) s += dot4(h4[j],   w[Hs / 4 + j]);
            sc[tid] = s; red[tid] = s;
        }
        __syncthreads();
        for (int s = 128; s > 0; s >>= 1) {
            if (tid < s) red[tid] = fmaxf(red[tid], red[tid + s]);
            __syncthreads();
        }
        const float mx = red[0];
        __syncthreads();
        if (tid < Ts) { const float e = __expf(sc[tid] - mx); sc[tid] = e; red[tid] = e; }
        __syncthreads();
        for (int s = 128; s > 0; s >>= 1) {
            if (tid < s) red[tid] += red[tid + s];
            __syncthreads();
        }
        const float inv = 1.0f / red[0];
        __syncthreads();
        if (tid < Ts) sc[tid] *= inv;
        __syncthreads();

        // context: ctx[d] = sum_j aw[j] * enc_outs[j][d]  (coalesced across tid)
        {
            float c = 0.0f;
            for (int j = 0; j < Ts; ++j) c += sc[j] * enc_outs[(size_t)j * Hs + tid];
            ctx[tid] = c;
        }
        __syncthreads();

        // comb[d] = relu(concat(emb,ctx) . comb_W[d] + comb_b[d])
        {
            const float4* w = (const float4*)(comb_W + (size_t)tid * (2 * Hs));
            float s = comb_b[tid];
            for (int j = 0; j < Hs / 4; ++j) s += dot4(emb4[j], w[j]);
            for (int j = 0; j < Hs / 4; ++j) s += dot4(ctx4[j], w[Hs / 4 + j]);
            comb[tid] = fmaxf(s, 0.0f);
        }
        __syncthreads();

        // GRU cell: thread tid forms all 6 dots for its hidden dim
        float xr = bxr, xz = bxz, xn = bxn;
        float hr = bhr, hz = bhz, hn = bhn;
        for (int j = 0; j < Hs / 4; ++j) {
            const float4 cj = comb4[j], hj = h4[j];
            xr += dot4(cj, wir[j]);  xz += dot4(cj, wiz[j]);  xn += dot4(cj, win[j]);
            hr += dot4(hj, whr[j]);  hz += dot4(hj, whz[j]);  hn += dot4(hj, whn[j]);
        }
        const float r = sigm(xr + hr);
        const float z = sigm(xz + hz);
        const float n = tanhf(xn + r * hn);
        const float hnew = (1.0f - z) * n + z * h[tid];
        __syncthreads();
        h[tid] = hnew;
        Hdec[(size_t)t * Hs + tid] = hnew;
        __syncthreads();
    }
}

// ---------------------------------------------------------------------------
// In-place row-wise log-softmax over V: one block per row.
// ---------------------------------------------------------------------------
__global__ void __launch_bounds__(256)
log_softmax_rows(float* __restrict__ X, int N)
{
    __shared__ float red[256];
    const int row = blockIdx.x;
    const int tid = threadIdx.x;
    float* x = X + (size_t)row * N;

    float m = -INFINITY;
    for (int i = tid; i < N; i += 256) m = fmaxf(m, x[i]);
    red[tid] = m; __syncthreads();
    for (int s = 128; s > 0; s >>= 1) {
        if (tid < s) red[tid] = fmaxf(red[tid], red[tid + s]);
        __syncthreads();
    }
    m = red[0]; __syncthreads();

    float sum = 0.0f;
    for (int i = tid; i < N; i += 256) sum += __expf(x[i] - m);
    red[tid] = sum; __syncthreads();
    for (int s = 128; s > 0; s >>= 1) {
        if (tid < s) red[tid] += red[tid + s];
        __syncthreads();
    }
    const float lse = m + __logf(red[0]);
    __syncthreads();

    for (int i = tid; i < N; i += 256) x[i] = x[i] - lse;
}

// ---------------------------------------------------------------------------
extern "C" void kernel_launch(void* const* d_in, const int* in_sizes, int n_in,
                              void* d_out, int out_size, void* d_ws, size_t ws_size,
                              hipStream_t stream)
{
    const float* input_seq = (const float*)d_in[0];   // [T,1,H]
    const int*   target    = (const int*)  d_in[1];   // [T,1]
    const float* emb_dec   = (const float*)d_in[2];   // [V,H]
    const float* enc_Wih   = (const float*)d_in[3];   // [3H,H]
    const float* enc_Whh   = (const float*)d_in[4];
    const float* enc_bih   = (const float*)d_in[5];
    const float* enc_bhh   = (const float*)d_in[6];
    const float* attn_W    = (const float*)d_in[7];   // [T,2H]
    const float* attn_b    = (const float*)d_in[8];
    const float* comb_W    = (const float*)d_in[9];   // [H,2H]
    const float* comb_b    = (const float*)d_in[10];
    const float* dec_Wih   = (const float*)d_in[11];
    const float* dec_Whh   = (const float*)d_in[12];
    const float* dec_bih   = (const float*)d_in[13];
    const float* dec_bhh   = (const float*)d_in[14];
    const float* out_W     = (const float*)d_in[15];  // [V,H]
    const float* out_b     = (const float*)d_in[16];
    float* out = (float*)d_out;                       // [T,1,V] log-probs

    float* gx_enc   = (float*)d_ws;                   // T*3H
    float* enc_outs = gx_enc + (size_t)Ts * H3;       // T*H
    float* Hdec     = enc_outs + (size_t)Ts * Hs;     // T*H

    // 1) batched encoder input projection (WMMA):  gx_enc = input_seq @ Wih^T + bih
    {
        dim3 blk(256);
        dim3 grd((H3 / 16 + 7) / 8, Ts / (16 * MTILE));
        wmma_gemm_bias_f32<<<grd, blk, 0, stream>>>(input_seq, enc_Wih, enc_bih,
                                                    gx_enc, Ts, H3, Hs);
    }

    // 2) sequential encoder GRU scan
    encoder_scan<<<1, 1024, 0, stream>>>(gx_enc, enc_Whh, enc_bhh, enc_outs);

    // 3) sequential decoder scan (attention + GRU), emits Hdec
    decoder_scan<<<1, 1024, 0, stream>>>(target, emb_dec, attn_W, attn_b,
                                         comb_W, comb_b,
                                         dec_Wih, dec_bih, dec_Whh, dec_bhh,
                                         enc_outs, Hdec);

    // 4) deferred batched vocab projection (WMMA): logits = Hdec @ out_W^T + out_b
    {
        const int ntiles = (Vs + 15) / 16;            // 3142
        dim3 blk(256);
        dim3 grd((ntiles + 7) / 8, Ts / (16 * MTILE));
        wmma_gemm_bias_f32<<<grd, blk, 0, stream>>>(Hdec, out_W, out_b,
                                                    out, Ts, Vs, Hs);
    }

    // 5) in-place log-softmax per row
    log_softmax_rows<<<Ts, 256, 0, stream>>>(out, Vs);
}